// LSTM_71554155152142
// MI455X (gfx1250) — compile-verified
//
#include <hip/hip_runtime.h>

// Problem constants (from reference)
#define NN      1000          // series
#define EE      8             // env features
#define HH      2048          // hidden
#define TT      24            // timesteps
#define IN_DIM  12000         // N*(E+4)
#define KNOWN   11000         // coords(2000)+env(8000)+dusk(1000): non-recurrent part
#define KPAD    11008         // KNOWN padded to multiple of 4 (zeros)
#define G4H     8192          // 4*H gate rows
#define TCOLS   32            // 24 timesteps padded to 32 (cols 24..31 unused)
#define KCHUNKS 4
#define KCHUNK_LEN 2752       // KPAD / KCHUNKS

typedef __attribute__((ext_vector_type(2))) float v2f;
typedef __attribute__((ext_vector_type(8))) float v8f;

__device__ __forceinline__ float sigmoidf_(float v) { return 1.0f / (1.0f + expf(-v)); }

// ---------------------------------------------------------------------------
// Build packed known-input matrix Kpk.
// Logical K[t][j], t in [0,32), j in [0,11008): rows t>=24 and cols j>=11000 are 0.
// Packed layout for single-v2f WMMA B-fragment loads:
//   Kpk[(j>>1)*64 + t*2 + (j&1)]
// so lane n reads {K[t=n][k], K[t=n][k+1]} as one 8-byte load (k even).
// ---------------------------------------------------------------------------
__global__ void buildK_kernel(const float* __restrict__ coords,
                              const float* __restrict__ env,
                              const float* __restrict__ dusk,
                              float* __restrict__ Kpk) {
    int tid = blockIdx.x * 256 + threadIdx.x;       // 0 .. 352255
    int j = tid >> 5;
    int t = tid & 31;
    float v = 0.0f;
    if (t < TT && j < KNOWN) {
        if (j < 2 * NN) {
            v = coords[j];                                   // coords.reshape(-1)
        } else if (j < 2 * NN + NN * EE) {
            int q = j - 2 * NN;                              // env_t flat (N,E)
            int n = q >> 3, e = q & 7;
            v = env[(size_t)(n * EE + e) * (TT + 1) + (t + 1)];
        } else {
            int n = j - (2 * NN + NN * EE);                  // dusk_t
            v = dusk[(size_t)n * TT + t];
        }
    }
    Kpk[(size_t)(j >> 1) * 64 + t * 2 + (j & 1)] = v;
}

// h=0, c=0, x0 = x[:,0]; also write output column 0.
__global__ void initState_kernel(const float* __restrict__ x,
                                 float* __restrict__ hA, float* __restrict__ cb,
                                 float* __restrict__ xA, float* __restrict__ out) {
    int i = blockIdx.x * 256 + threadIdx.x;
    if (i < HH) { hA[i] = 0.0f; cb[i] = 0.0f; }
    if (i < NN) {
        float v = x[(size_t)i * (TT + 1)];
        xA[i] = v;
        out[(size_t)i * (TT + 1)] = v;
    }
}

// ---------------------------------------------------------------------------
// Z-partial GEMM with V_WMMA_F32_16X16X4_F32:
//   Zpart[kchunk][row][tcol] = sum_{j in chunk} W_ih[row][j] * K[tcol][j]
// M = 8192 (16-row tiles), N = 32 (two 16-col tiles sharing one A fragment),
// K = 11008 split into 4 chunks -> 2048 waves, memory-bound on the 360 MB
// W_ih[:, :11000] stream (~15 us floor at 23.3 TB/s HBM).
// The W stream is read-once: non-temporal loads keep it from evicting the
// recurrent weights (W_hh + W_ih x-columns, ~100 MB) that the 24 sequential
// steps want resident in the 192 MB L2. Zpart is write-once -> NT stores.
// A frag: lanes 0-15 -> rows m0..m0+15 k={0,1}; lanes 16-31 -> k={2,3}.
// B frag: lanes 0-15 -> cols 0..15   k={0,1}; lanes 16-31 -> k={2,3}.
// ---------------------------------------------------------------------------
__global__ void gemm_known_kernel(const float* __restrict__ Wih,
                                  const float* __restrict__ Kpk,
                                  float* __restrict__ Zpart) {
    int wave   = blockIdx.x * 8 + (threadIdx.x >> 5);  // 0..2047
    int lane   = threadIdx.x & 31;
    int mtile  = wave & 511;
    int kchunk = wave >> 9;
    int m0     = mtile << 4;
    int half   = lane >> 4;        // 0: k 0,1   1: k 2,3
    int l15    = lane & 15;

    const float* Arow = Wih + (size_t)(m0 + l15) * IN_DIM + half * 2;

    v8f c0 = {0.f,0.f,0.f,0.f,0.f,0.f,0.f,0.f};
    v8f c1 = {0.f,0.f,0.f,0.f,0.f,0.f,0.f,0.f};

    int kbeg = kchunk * KCHUNK_LEN;
    int kend = kbeg + KCHUNK_LEN;
#pragma unroll 4
    for (int kk = kbeg; kk < kend; kk += 4) {
        v2f a = __builtin_nontemporal_load((const v2f*)(Arow + kk));   // read-once stream
        const float* bp = Kpk + (size_t)((kk + half * 2) >> 1) * 64;   // tiny, reused: RT
        v2f b0 = *(const v2f*)(bp + l15 * 2);                          // cols 0..15
        v2f b1 = *(const v2f*)(bp + (16 + l15) * 2);                   // cols 16..31
        c0 = __builtin_amdgcn_wmma_f32_16x16x4_f32(false, a, false, b0,
                                                   (short)0, c0, false, false);
        c1 = __builtin_amdgcn_wmma_f32_16x16x4_f32(false, a, false, b1,
                                                   (short)0, c1, false, false);
    }

    // C/D layout: VGPR r, lanes 0-15 -> M=r, N=lane; lanes 16-31 -> M=r+8, N=lane-16
    size_t base = (size_t)kchunk * G4H;
#pragma unroll
    for (int r = 0; r < 8; ++r) {
        int row = m0 + r + half * 8;
        __builtin_nontemporal_store(c0[r], &Zpart[(base + row) * TCOLS + l15]);
        __builtin_nontemporal_store(c1[r], &Zpart[(base + row) * TCOLS + 16 + l15]);
    }
}

// Z[g][t] = sum_chunks Zpart + b_ih[g] + b_hh[g]   (deterministic fixed-order sum)
__global__ void reduceZ_kernel(const float* __restrict__ Zpart,
                               const float* __restrict__ bih,
                               const float* __restrict__ bhh,
                               float* __restrict__ Z) {
    int idx = blockIdx.x * 256 + threadIdx.x;    // 0 .. 8192*32-1
    int g = idx >> 5;
    float s = bih[g] + bhh[g];
#pragma unroll
    for (int c = 0; c < KCHUNKS; ++c)
        s += __builtin_nontemporal_load(&Zpart[(size_t)c * G4H * TCOLS + idx]);
    Z[idx] = s;
}

__device__ __forceinline__ float dot4acc(float4 w, float4 v, float a) {
    return fmaf(w.x, v.x, fmaf(w.y, v.y, fmaf(w.z, v.z, fmaf(w.w, v.w, a))));
}

// ---------------------------------------------------------------------------
// One recurrent step: wave k computes gate rows {k, k+H, k+2H, k+3H} as
// dot products over recurrent-x (1000, W_ih cols 11000..) and h (2048, W_hh),
// wave32 shuffle-reduces, then the LSTM cell update. float4 lanes -> b128
// loads (512 B per load instruction) so the ~100 MB/step L2-resident stream
// is bandwidth-limited, not VMEM-issue-limited. All bases 16 B aligned.
// ---------------------------------------------------------------------------
__global__ void lstm_step_kernel(int t,
                                 const float* __restrict__ Wih,
                                 const float* __restrict__ Whh,
                                 const float* __restrict__ Z,
                                 const float* __restrict__ xcur,
                                 const float* __restrict__ hold,
                                 float* __restrict__ hnew,
                                 float* __restrict__ cb) {
    int k    = blockIdx.x * 8 + (threadIdx.x >> 5);   // hidden unit, 0..2047
    int lane = threadIdx.x & 31;

    const float4* w0 = (const float4*)(Wih + (size_t)(k         ) * IN_DIM + KNOWN);
    const float4* w1 = (const float4*)(Wih + (size_t)(k +     HH) * IN_DIM + KNOWN);
    const float4* w2 = (const float4*)(Wih + (size_t)(k + 2 * HH) * IN_DIM + KNOWN);
    const float4* w3 = (const float4*)(Wih + (size_t)(k + 3 * HH) * IN_DIM + KNOWN);
    const float4* xv4 = (const float4*)xcur;

    float a0 = 0.f, a1 = 0.f, a2 = 0.f, a3 = 0.f;
    for (int j = lane; j < NN / 4; j += 32) {        // 250 float4s
        float4 xv = xv4[j];
        a0 = dot4acc(w0[j], xv, a0);
        a1 = dot4acc(w1[j], xv, a1);
        a2 = dot4acc(w2[j], xv, a2);
        a3 = dot4acc(w3[j], xv, a3);
    }

    const float4* u0 = (const float4*)(Whh + (size_t)(k         ) * HH);
    const float4* u1 = (const float4*)(Whh + (size_t)(k +     HH) * HH);
    const float4* u2 = (const float4*)(Whh + (size_t)(k + 2 * HH) * HH);
    const float4* u3 = (const float4*)(Whh + (size_t)(k + 3 * HH) * HH);
    const float4* hv4 = (const float4*)hold;
    for (int j = lane; j < HH / 4; j += 32) {        // 512 float4s
        float4 hv = hv4[j];
        a0 = dot4acc(u0[j], hv, a0);
        a1 = dot4acc(u1[j], hv, a1);
        a2 = dot4acc(u2[j], hv, a2);
        a3 = dot4acc(u3[j], hv, a3);
    }

    for (int off = 16; off; off >>= 1) {
        a0 += __shfl_xor(a0, off, 32);
        a1 += __shfl_xor(a1, off, 32);
        a2 += __shfl_xor(a2, off, 32);
        a3 += __shfl_xor(a3, off, 32);
    }

    if (lane == 0) {
        float gi = Z[(size_t)(k         ) * TCOLS + t] + a0;
        float gf = Z[(size_t)(k +     HH) * TCOLS + t] + a1;
        float gg = Z[(size_t)(k + 2 * HH) * TCOLS + t] + a2;
        float go = Z[(size_t)(k + 3 * HH) * TCOLS + t] + a3;
        float cold = cb[k];
        float cn = sigmoidf_(gf) * cold + sigmoidf_(gi) * tanhf(gg);
        float hn = sigmoidf_(go) * tanhf(cn);
        cb[k]   = cn;     // per-k private -> in-place is safe
        hnew[k] = hn;     // double-buffered vs hold
    }
}

// x_new[n] = sigmoid(W_out[n,:] @ h_new + b_out[n]) * night[n][t+1]; one wave per n.
__global__ void out_step_kernel(int t,
                                const float* __restrict__ Wout,
                                const float* __restrict__ bout,
                                const float* __restrict__ night,
                                const float* __restrict__ hnew,
                                float* __restrict__ xnext,
                                float* __restrict__ out) {
    int n    = blockIdx.x * 8 + (threadIdx.x >> 5);   // 0..999 (grid exact)
    int lane = threadIdx.x & 31;
    const float4* w   = (const float4*)(Wout + (size_t)n * HH);
    const float4* hv4 = (const float4*)hnew;
    float a = 0.f;
    for (int j = lane; j < HH / 4; j += 32) a = dot4acc(w[j], hv4[j], a);
    for (int off = 16; off; off >>= 1) a += __shfl_xor(a, off, 32);
    if (lane == 0) {
        float v = sigmoidf_(a + bout[n]) * night[(size_t)n * (TT + 1) + t + 1];
        xnext[n] = v;
        out[(size_t)n * (TT + 1) + t + 1] = v;
    }
}

extern "C" void kernel_launch(void* const* d_in, const int* in_sizes, int n_in,
                              void* d_out, int out_size, void* d_ws, size_t ws_size,
                              hipStream_t stream) {
    const float* x      = (const float*)d_in[0];
    const float* coords = (const float*)d_in[1];
    const float* env    = (const float*)d_in[2];
    const float* dusk   = (const float*)d_in[3];
    const float* night  = (const float*)d_in[4];
    const float* Wih    = (const float*)d_in[5];
    const float* bih    = (const float*)d_in[6];
    const float* Whh    = (const float*)d_in[7];
    const float* bhh    = (const float*)d_in[8];
    const float* Wout   = (const float*)d_in[9];
    const float* bout   = (const float*)d_in[10];
    float* out = (float*)d_out;

    // Workspace carve-up (floats): ~6.7 MB total
    float* ws    = (float*)d_ws;
    float* Kpk   = ws;                         // (11008/2)*64        = 352256
    float* Zpart = Kpk   + 352256;             // 4*8192*32           = 1048576
    float* Z     = Zpart + 1048576;            // 8192*32             = 262144
    float* hA    = Z     + 262144;             // 2048
    float* hB    = hA    + 2048;               // 2048
    float* cb    = hB    + 2048;               // 2048
    float* xA    = cb    + 2048;               // 1024
    float* xB    = xA    + 1024;               // 1024

    buildK_kernel   <<<1376, 256, 0, stream>>>(coords, env, dusk, Kpk);
    initState_kernel<<<8,    256, 0, stream>>>(x, hA, cb, xA, out);
    gemm_known_kernel<<<256, 256, 0, stream>>>(Wih, Kpk, Zpart);
    reduceZ_kernel  <<<1024, 256, 0, stream>>>(Zpart, bih, bhh, Z);

    for (int t = 0; t < TT; ++t) {
        const float* xcur  = (t & 1) ? xB : xA;
        float*       xnext = (t & 1) ? xA : xB;
        const float* hold  = (t & 1) ? hB : hA;
        float*       hnew  = (t & 1) ? hA : hB;
        lstm_step_kernel<<<256, 256, 0, stream>>>(t, Wih, Whh, Z, xcur, hold, hnew, cb);
        out_step_kernel <<<125, 256, 0, stream>>>(t, Wout, bout, night, hnew, xnext, out);
    }
}